// ExampleTiedDropout_65214783422518
// MI455X (gfx1250) — compile-verified
//
#include <hip/hip_runtime.h>
#include <stdint.h>

// MI455X / gfx1250, wave32. Pure HBM-streaming op (no matmul structure ->
// WMMA/TDM intentionally unused; bandwidth-bound at 23.3 TB/s).
// Layout: ONE ROW (2048 f32 = 8 KB) PER WAVE so the wave-uniform Threefry
// mask hash (scalarized to the SALU pipe) runs once per row, not once per
// wave-slice: 32768 waves x ~300 SALU ~= 10M scalar ops (~5 us aggregate),
// below the ~15 us memory floor. Zero rows (~72%) skip all reads.

typedef __attribute__((ext_vector_type(4))) float v4f;

// Matches JAX's threefry_partitionable default (True since JAX 0.4.36).
// Set to 0 for the legacy pre-partitionable random_bits layout.
#define JAX_THREEFRY_PARTITIONABLE 1

// ---- Threefry-2x32, 20 rounds; bit-exact with jax._src.prng.threefry2x32 ----
__device__ __forceinline__ void tf2x32(uint32_t k0, uint32_t k1,
                                       uint32_t& x0, uint32_t& x1) {
  const uint32_t k2 = k0 ^ k1 ^ 0x1BD11BDAu;
#define TFR(r) { x0 += x1; x1 = (x1 << (r)) | (x1 >> (32 - (r))); x1 ^= x0; }
  x0 += k0; x1 += k1;
  TFR(13) TFR(15) TFR(26) TFR(6)   x0 += k1; x1 += k2 + 1u;
  TFR(17) TFR(29) TFR(16) TFR(24)  x0 += k2; x1 += k0 + 2u;
  TFR(13) TFR(15) TFR(26) TFR(6)   x0 += k0; x1 += k1 + 3u;
  TFR(17) TFR(29) TFR(16) TFR(24)  x0 += k1; x1 += k2 + 4u;
  TFR(13) TFR(15) TFR(26) TFR(6)   x0 += k2; x1 += k0 + 5u;
#undef TFR
}

__device__ __forceinline__ float row_mask(const int* __restrict__ idx,
                                          int b, int s) {
  constexpr int NFIX = 204;   // int(0.2 * 1024)
  if (s < NFIX) return 1.0f;
  // key_b = fold_in(key(0), idx[b]) = threefry((0,0), (0, uint32(idx[b])))
  uint32_t c0 = 0u, c1 = (uint32_t)idx[b];
  tf2x32(0u, 0u, c0, c1);
  const int j = s - NFIX;                 // position within the 820 random slots
#if JAX_THREEFRY_PARTITIONABLE
  // random_bits (partitionable): counts = 64-bit iota split (hi=0, lo=j);
  // 32-bit bits = out0 ^ out1.
  uint32_t x0 = 0u, x1 = (uint32_t)j;
  tf2x32(c0, c1, x0, x1);
  const uint32_t bits = x0 ^ x1;
#else
  // legacy random_bits: iota(820) split in half -> pair (p, p+410),
  // first half takes out0, second half takes out1.
  constexpr int HALF = 410;
  const int p = (j < HALF) ? j : j - HALF;
  uint32_t x0 = (uint32_t)p, x1 = (uint32_t)(p + HALF);
  tf2x32(c0, c1, x0, x1);
  const uint32_t bits = (j < HALF) ? x0 : x1;
#endif
  // uniform[0,1): bitcast((bits>>9)|0x3F800000) - 1.0 ; keep iff u < 0.1f
  const float u = __uint_as_float((bits >> 9) | 0x3F800000u) - 1.0f;
  return (u < 0.1f) ? 1.0f : 0.0f;
}

__global__ __launch_bounds__(256, 1)
void TiedDropoutStream_gfx1250(const v4f* __restrict__ X,
                               const int* __restrict__ idx,
                               v4f* __restrict__ out) {
  constexpr int SEQ = 1024;
  constexpr int VEC_PER_ROW = 2048 / 4;   // 512 float4 per row
  constexpr int CHUNKS = VEC_PER_ROW / 32; // 16 b128 ops per lane per row

  const int lane = (int)threadIdx.x & 31;
  // One row per wave32. Force the row id into an SGPR so the Threefry hash
  // and the mask branch stay on the scalar pipe (threadIdx.x>>5 is wave-
  // uniform but not provably so to the compiler).
  int row = ((int)blockIdx.x << 3) | ((int)threadIdx.x >> 5);
  row = __builtin_amdgcn_readfirstlane(row);

  const int s = row & (SEQ - 1);
  const int b = row >> 10;

  // Wave-uniform: SALU threefry, s_load of idx[b], s_cbranch on the result.
  const float m = row_mask(idx, b, s);

  const size_t base = (size_t)row * VEC_PER_ROW + lane;
  if (m != 0.0f) {
    // touch-once streams -> non-temporal b128 (don't churn the 192 MB L2);
    // each instruction moves a contiguous 512 B per wave.
#pragma unroll
    for (int k = 0; k < CHUNKS; ++k) {
      v4f a = __builtin_nontemporal_load(&X[base + 32 * k]);
      a *= m;
      __builtin_nontemporal_store(a, &out[base + 32 * k]);
    }
  } else {
    // ~72% of rows: skip the 8 KB read entirely, just stream zeros.
    const v4f z = {0.0f, 0.0f, 0.0f, 0.0f};
#pragma unroll
    for (int k = 0; k < CHUNKS; ++k)
      __builtin_nontemporal_store(z, &out[base + 32 * k]);
  }
}

extern "C" void kernel_launch(void* const* d_in, const int* in_sizes, int n_in,
                              void* d_out, int out_size, void* d_ws, size_t ws_size,
                              hipStream_t stream) {
  (void)n_in; (void)out_size; (void)d_ws; (void)ws_size;
  const v4f* X   = (const v4f*)d_in[0];     // [32,1024,2048] f32
  const int* idx = (const int*)d_in[1];     // [32] i32
  v4f* out       = (v4f*)d_out;             // [32,1024,2048] f32

  const int rows   = in_sizes[0] / 2048;    // 32 * 1024 = 32768
  const int blocks = rows / 8;              // 8 rows (waves) per 256-thread block
  TiedDropoutStream_gfx1250<<<blocks, 256, 0, stream>>>(X, idx, out);
}